// TryLoss_49959059587490
// MI455X (gfx1250) — compile-verified
//
#include <hip/hip_runtime.h>
#include <hip/hip_bf16.h>
#include <stdint.h>

// Problem constants (from reference)
#define NVV      778
#define NSEG     16
#define SEGSZ    49
#define GRIDN    32
#define BATCH    128
#define NPTS     (NSEG * SEGSZ)          // 784
#define VOLN     (GRIDN * GRIDN * GRIDN) // 32768 floats = 128KB
#define NVF      (NVV * 3)               // 2334 floats of vertex data
#define THREADS  256

#if defined(__HIP_DEVICE_COMPILE__)
#if __has_builtin(__builtin_amdgcn_global_load_async_to_lds_b128) && \
    __has_builtin(__builtin_amdgcn_s_wait_asynccnt)
#define USE_ASYNC 1
#endif
#if __has_builtin(__builtin_amdgcn_wmma_f32_16x16x4_f32)
#define USE_WMMA 1
#endif
#endif

typedef __attribute__((ext_vector_type(2))) float v2f;
typedef __attribute__((ext_vector_type(8))) float v8f;
typedef __attribute__((ext_vector_type(4))) int   v4i;

#ifdef USE_ASYNC
typedef __attribute__((address_space(1))) v4i gv4i;   // global int4
typedef __attribute__((address_space(3))) v4i lv4i;   // LDS int4
#endif

__device__ __forceinline__ int clamp31(int v) {
    return v < 0 ? 0 : (v > GRIDN - 1 ? GRIDN - 1 : v);
}
__device__ __forceinline__ float maskw(int v, float w) {
    // weight if 0 <= v < GRIDN else 0 (zeros padding)
    return ((unsigned)v < (unsigned)GRIDN) ? w : 0.0f;
}

// One block per (b, hand, j):  blk = b*32 + hand*16 + j
// hand==0: sample LEFT verts (vertices[:,1]) in RIGHT boxes on phi_right
// hand==1: sample RIGHT verts (vertices[:,0]) in LEFT boxes on phi_left
__global__ __launch_bounds__(THREADS) void sample_partial_kernel(
    const float* __restrict__ vertices, const float* __restrict__ phiR,
    const float* __restrict__ phiL, const int* __restrict__ seg,
    float* __restrict__ partial)
{
    __shared__ float vol_s[VOLN];        // 128KB volume tile
    __shared__ float verts_s[NVF];       // src-hand vertices, coalesced copy
    __shared__ int   seg_s[NPTS];
    __shared__ float bxs[SEGSZ], bys[SEGSZ], bzs[SEGSZ];
    __shared__ float s_cx, s_cy, s_cz, s_inv;
    __shared__ float s_wsum[8];

    const int tid = threadIdx.x;
    const int blk = blockIdx.x;
    const int b   = blk >> 5;
    const int r   = blk & 31;
    const int h   = r >> 4;
    const int j   = r & 15;

    const int hsrc = (h == 0) ? 1 : 0;   // vertices[:,1] is the left hand
    const int hbox = 1 - hsrc;
    const float* vsrc = vertices + ((size_t)(b * 2 + hsrc)) * NVF;
    const float* vbox = vertices + ((size_t)(b * 2 + hbox)) * NVF;
    const float* phi  = (h == 0) ? phiR : phiL;
    const float* vol  = phi + ((size_t)(j * BATCH + b)) * VOLN;

    // ---- 1) kick off async volume copy global -> LDS (ASYNCcnt path) ----
#ifdef USE_ASYNC
    {
        gv4i* g = (gv4i*)(uintptr_t)vol;                          // 8192 int4's
        lv4i* l = (lv4i*)(uint32_t)(uintptr_t)(void*)vol_s;
        #pragma unroll 8
        for (int it = 0; it < VOLN / 4 / THREADS; ++it) {         // 32 iters
            const int idx = it * THREADS + tid;                   // int4 index
            __builtin_amdgcn_global_load_async_to_lds_b128(g + idx, l + idx, 0, 0);
        }
    }
#else
    for (int it = 0; it < VOLN / 4 / THREADS; ++it) {
        const int idx = it * THREADS + tid;
        ((float4*)vol_s)[idx] = ((const float4*)vol)[idx];
    }
#endif

    // ---- 2) stream src-hand vertices + seg into LDS (coalesced) ----
    for (int i = tid; i < NVF; i += THREADS) verts_s[i] = vsrc[i];
    for (int i = tid; i < NPTS; i += THREADS) seg_s[i] = seg[i];

    // ---- 3) box-defining vertices of segment j (opposite hand) ----
    if (tid < SEGSZ) {
        const int vid = seg[j * SEGSZ + tid];
        bxs[tid] = vbox[vid * 3 + 0];
        bys[tid] = vbox[vid * 3 + 1];
        bzs[tid] = vbox[vid * 3 + 2];
    }
    __syncthreads();
    if (tid == 0) {
        float mnx = bxs[0], mxx = bxs[0];
        float mny = bys[0], mxy = bys[0];
        float mnz = bzs[0], mxz = bzs[0];
        for (int v = 1; v < SEGSZ; ++v) {
            mnx = fminf(mnx, bxs[v]); mxx = fmaxf(mxx, bxs[v]);
            mny = fminf(mny, bys[v]); mxy = fmaxf(mxy, bys[v]);
            mnz = fminf(mnz, bzs[v]); mxz = fmaxf(mxz, bzs[v]);
        }
        s_cx = 0.5f * (mnx + mxx);
        s_cy = 0.5f * (mny + mxy);
        s_cz = 0.5f * (mnz + mxz);
        const float ext   = fmaxf(mxx - mnx, fmaxf(mxy - mny, mxz - mnz));
        const float scale = 1.1f * 0.5f * ext;     // (1+SCALE_FACTOR)*0.5*ext
        s_inv = 1.0f / scale;
    }
#ifdef USE_ASYNC
    __builtin_amdgcn_s_wait_asynccnt(0);   // own copies done; barrier covers other waves
#endif
    __syncthreads();

    // ---- 4) trilinear sampling from LDS: straight-line, masked weights ----
    const float GS = 0.5f * (float)(GRIDN - 1);    // 15.5
    const float cx = s_cx, cy = s_cy, cz = s_cz;
    const float a  = s_inv * GS;                   // ((p-c)*inv + 1)*GS = (p-c)*a + GS
    float acc = 0.0f;
    for (int p = tid; p < NPTS; p += THREADS) {
        const int vid = seg_s[p];
        const float fx = (verts_s[vid * 3 + 0] - cx) * a + GS;
        const float fy = (verts_s[vid * 3 + 1] - cy) * a + GS;
        const float fz = (verts_s[vid * 3 + 2] - cz) * a + GS;
        const float x0 = floorf(fx), y0 = floorf(fy), z0 = floorf(fz);
        const int ix = (int)x0, iy = (int)y0, iz = (int)z0;
        const float wx = fx - x0, wy = fy - y0, wz = fz - z0;

        const float wx0 = maskw(ix,     1.0f - wx), wx1 = maskw(ix + 1, wx);
        const float wy0 = maskw(iy,     1.0f - wy), wy1 = maskw(iy + 1, wy);
        const float wz0 = maskw(iz,     1.0f - wz), wz1 = maskw(iz + 1, wz);

        const int cx0 = clamp31(ix), cx1 = clamp31(ix + 1);
        const int cy0 = clamp31(iy), cy1 = clamp31(iy + 1);
        const int cz0 = clamp31(iz), cz1 = clamp31(iz + 1);

        const int b00 = (cz0 * GRIDN + cy0) * GRIDN;
        const int b01 = (cz0 * GRIDN + cy1) * GRIDN;
        const int b10 = (cz1 * GRIDN + cy0) * GRIDN;
        const int b11 = (cz1 * GRIDN + cy1) * GRIDN;

        const float v000 = vol_s[b00 + cx0], v001 = vol_s[b00 + cx1];
        const float v010 = vol_s[b01 + cx0], v011 = vol_s[b01 + cx1];
        const float v100 = vol_s[b10 + cx0], v101 = vol_s[b10 + cx1];
        const float v110 = vol_s[b11 + cx0], v111 = vol_s[b11 + cx1];

        acc += wz0 * (wy0 * (wx0 * v000 + wx1 * v001) +
                      wy1 * (wx0 * v010 + wx1 * v011)) +
               wz1 * (wy0 * (wx0 * v100 + wx1 * v101) +
                      wy1 * (wx0 * v110 + wx1 * v111));
    }

    // ---- 5) wave reduction via V_WMMA_F32_16X16X4_F32 (B = ones) ----
#ifdef USE_WMMA
    {
        // A layout (16x4 f32): VGPR0 = K0 (lanes 0-15) / K2 (lanes 16-31);
        // VGPR1 = K1/K3. With A={acc,0}, D[m,n] = acc[m] + acc[m+16].
        v2f av; av[0] = acc; av[1] = 0.0f;
        v2f bm; bm[0] = 1.0f; bm[1] = 1.0f;
        v8f c = {};
        c = __builtin_amdgcn_wmma_f32_16x16x4_f32(
                false, av, false, bm, (short)0, c, false, false);
        // D VGPR r: lanes0-15 hold rowsum(r), lanes16-31 hold rowsum(8+r)
        float t = c[0] + c[1] + c[2] + c[3] + c[4] + c[5] + c[6] + c[7];
        const float ws = __shfl(t, 0, 32) + __shfl(t, 16, 32);
        if ((tid & 31) == 0) s_wsum[tid >> 5] = ws;
    }
    __syncthreads();
    if (tid == 0) {
        float s = 0.0f;
        for (int w = 0; w < 8; ++w) s += s_wsum[w];
        partial[blk] = s;
    }
#else
    __shared__ float red[THREADS];
    red[tid] = acc;
    __syncthreads();
    for (int off = THREADS / 2; off > 0; off >>= 1) {
        if (tid < off) red[tid] += red[tid + off];
        __syncthreads();
    }
    if (tid == 0) partial[blk] = red[0];
#endif
}

// Deterministic final reduction: loss[b] = 0.25 * sum_{r<32} partial[b*32+r]
__global__ void reduce_loss_kernel(const float* __restrict__ partial,
                                   float* __restrict__ out)
{
    const int b = threadIdx.x;
    if (b < BATCH) {
        float s = 0.0f;
        for (int r = 0; r < 32; ++r) s += partial[b * 32 + r];
        out[b] = 0.25f * s;
    }
}

extern "C" void kernel_launch(void* const* d_in, const int* in_sizes, int n_in,
                              void* d_out, int out_size, void* d_ws, size_t ws_size,
                              hipStream_t stream) {
    const float* vertices = (const float*)d_in[0];   // [128,2,778,3]
    const float* phiR     = (const float*)d_in[1];   // [16,128,32,32,32]
    const float* phiL     = (const float*)d_in[2];   // [16,128,32,32,32]
    const int*   seg      = (const int*)d_in[3];     // [16,49]
    float* partial = (float*)d_ws;                   // 4096 floats

    sample_partial_kernel<<<BATCH * 32, THREADS, 0, stream>>>(
        vertices, phiR, phiL, seg, partial);
    reduce_loss_kernel<<<1, BATCH, 0, stream>>>(partial, (float*)d_out);
}